// SpikeQuantizer_36558761624011
// MI455X (gfx1250) — compile-verified
//
#include <hip/hip_runtime.h>

// Integrate-and-fire spike quantizer (SNN ANN->SNN conversion), T=8, GROUP=128.
// Pure streaming kernel: 256 MB in + 256 MB out, touched once each.
// MI455X HBM roofline: 512 MB / 23.3 TB/s ~= 22 us. Arithmetic intensity ~1.1
// FLOP/byte -> memory bound; WMMA is inapplicable (fc1/fc2 are +/-1 sign splits).
//
// Grid layout avoids all integer division: x-dim tiles the feature dimension
// (group index = shift), y-dim strides over B*L rows.

#define SNN_T 8
#define SNN_GROUP 128
#define G4 (SNN_GROUP / 4)          // 32 float4 per quantization group

typedef float v4f __attribute__((ext_vector_type(4)));

__global__ __launch_bounds__(256) void snn_if_kernel(
    const float* __restrict__ x,
    const float* __restrict__ threshold,
    float* __restrict__ out,
    unsigned d4,        // d/4      : float4 columns per row
    unsigned rows,      // B*L      : rows per timestep plane
    unsigned plane4)    // d4*rows  : float4 per timestep plane
{
    unsigned col4 = blockIdx.x * blockDim.x + threadIdx.x;
    if (col4 >= d4) return;

    // One wave = 32 lanes * 4 floats = 128 elements = exactly one group:
    // threshold index is wave-uniform; G4 is a power of two -> pure shift.
    float th = threshold[col4 / G4];
    float th_half = 0.5f * th;

    for (unsigned row = blockIdx.y; row < rows; row += gridDim.y) {
        size_t i4 = (size_t)row * d4 + col4;
        const v4f* __restrict__ xp = (const v4f*)x + i4;
        v4f* __restrict__ op = (v4f*)out + i4;

        // The T loads are independent (only the IF scan is sequential):
        // issue all 8 nontemporal global_load_b128 up front for deep MLP.
        v4f xt[SNN_T];
#pragma unroll
        for (int t = 0; t < SNN_T; ++t)
            xt[t] = __builtin_nontemporal_load(xp + (size_t)t * plane4);

        // Membrane potentials pre-charged to th/2 (pos and neg pathways).
        float vp[4], vn[4];
#pragma unroll
        for (int k = 0; k < 4; ++k) { vp[k] = th_half; vn[k] = th_half; }

        // Soft-reset integrate-and-fire; out_t = (sp - sn)*th = dp - dn,
        // with dp,dn in {0, th} -> bit-exact vs reference.
#pragma unroll
        for (int t = 0; t < SNN_T; ++t) {
            v4f o;
#pragma unroll
            for (int k = 0; k < 4; ++k) {
                float xv = xt[t][k];
                vp[k] += xv;
                float dp = (vp[k] >= th) ? th : 0.0f;
                vp[k] -= dp;
                vn[k] -= xv;
                float dn = (vn[k] >= th) ? th : 0.0f;
                vn[k] -= dn;
                o[k] = dp - dn;
            }
            __builtin_nontemporal_store(o, op + (size_t)t * plane4);
        }
    }
}

extern "C" void kernel_launch(void* const* d_in, const int* in_sizes, int n_in,
                              void* d_out, int out_size, void* d_ws, size_t ws_size,
                              hipStream_t stream) {
    // inputs: x [T*B, L, d] f32; threshold [C,1,1] f32; fc1/fc2 params ignored
    // (weights pinned to +/-1, biases zero -> folded analytically).
    const float* x = (const float*)d_in[0];
    const float* threshold = (const float*)d_in[1];
    float* out = (float*)d_out;

    long long total = (long long)in_sizes[0];        // T*B*L*d
    unsigned C = (unsigned)in_sizes[1];              // number of groups
    unsigned d4 = C * G4;                            // float4 per row
    unsigned plane4 = (unsigned)(total / SNN_T / 4); // float4 per timestep plane
    unsigned rows = plane4 / d4;                     // B*L

    const unsigned block = 256;                      // 8 waves (wave32)
    unsigned gx = (d4 + block - 1) / block;          // tiles of feature dim
    unsigned gy = rows > 65535u ? 65535u : rows;     // rows (grid-stride if huge)

    snn_if_kernel<<<dim3(gx, gy), dim3(block), 0, stream>>>(
        x, threshold, out, d4, rows, plane4);
}